// Netsimple2_16226386444401
// MI455X (gfx1250) — compile-verified
//
#include <hip/hip_runtime.h>

// GraphConv x2 + Linear head for MI455X (gfx1250, wave32).
// Per layer: zero agg -> edge scatter (hw f32 atomics, L2-resident working set)
// -> fused node transform via V_WMMA_F32_16X16X4_F32:
//   out[16x16 tile] = [agg | x] (16x20) @ [W_rel ; W_root] (20x16) + b, ReLU.
// K=20 -> 5 accumulating K=4 WMMAs. Persistent waves hoist the B (weight)
// fragments; A-operand pairs never straddle buffers (k0 even, C even), so the
// hot loop is 5 unconditional float2 loads + 5 WMMAs + 8 immediate-offset
// stores. The (at most one) partial tail tile takes a separate guarded path.

#define C 10  // IN_C == HID == 10

typedef __attribute__((ext_vector_type(2))) float v2f;
typedef __attribute__((ext_vector_type(8))) float v8f;

// ---------------------------------------------------------------- zero ------
__global__ void zero_f32v4(float4* __restrict__ p, int n4) {
  int i = blockIdx.x * blockDim.x + threadIdx.x;
  if (i < n4) p[i] = make_float4(0.f, 0.f, 0.f, 0.f);
}

// ------------------------------------------------------------ edge scatter --
// One lane per edge: msg = ew * x[src]; atomic scatter-add into agg[dst].
// Rows are 40 B -> five 8 B (float2) loads, always 8-byte aligned.
// 10 global_atomic_add_f32 (no return -> STOREcnt, fire-and-forget into L2).
__global__ void edge_scatter(const float* __restrict__ x,
                             const int*   __restrict__ ei,  // [2, E]
                             const float* __restrict__ ew,
                             float* __restrict__ agg, int E) {
  int e = blockIdx.x * blockDim.x + threadIdx.x;
  if (e >= E) return;
  int   src = ei[e];
  int   dst = ei[E + e];
  float w   = ew[e];
  const float* xr = x   + (size_t)src * C;
  float*       ar = agg + (size_t)dst * C;
#pragma unroll
  for (int j = 0; j < 5; ++j) {
    float2 v = *reinterpret_cast<const float2*>(xr + 2 * j);
    __hip_atomic_fetch_add(ar + 2 * j + 0, w * v.x,
                           __ATOMIC_RELAXED, __HIP_MEMORY_SCOPE_AGENT);
    __hip_atomic_fetch_add(ar + 2 * j + 1, w * v.y,
                           __ATOMIC_RELAXED, __HIP_MEMORY_SCOPE_AGENT);
  }
}

// ----------------------------------------------------- fused node transform -
// A 16x4 f32 layout: lanes 0-15 hold M=0..15 with K={0,1} in the two VGPRs,
// lanes 16-31 hold K={2,3}; B mirrored with N = lane&15.
// D: VGPR r -> M = r + 8*(lane>=16), N = lane&15.
__device__ __forceinline__ void node_tile(
    bool guarded,                    // constant-folded after inlining
    int tile, int nNodes, int half, int nidx,
    const float* __restrict__ agg, const float* __restrict__ xin,
    const v2f* __restrict__ bf, float bv, float* __restrict__ xout) {
  int node = tile * 16 + nidx;
  if (guarded && node >= nNodes) node = nNodes - 1;  // clamp loads only
  const float* ra = agg + (size_t)node * C;
  const float* rx = xin + (size_t)node * C;

  v8f acc = {};
#pragma unroll
  for (int s = 0; s < 5; ++s) {
    const int k0 = 4 * s + 2 * half;
    // Pair (k0,k0+1) always lies within one buffer (k0 even, C even); the
    // ternary lowers to v_cndmask on the address + one aligned b64 load.
    const float* p = (k0 < C) ? (ra + k0) : (rx + (k0 - C));
    float2 av = *reinterpret_cast<const float2*>(p);
    v2f a;
    a.x = av.x;
    a.y = av.y;
    acc = __builtin_amdgcn_wmma_f32_16x16x4_f32(
        /*neg_a=*/false, a, /*neg_b=*/false, bf[s],
        /*c_mod=*/(short)0, acc, /*reuse_a=*/false, /*reuse_b=*/false);
  }

  if (nidx < C) {
    // Base pointer for this lane's column; rows r are immediate offsets r*40B.
    float* po = xout + (size_t)(tile * 16 + 8 * half) * C + nidx;
#pragma unroll
    for (int r = 0; r < 8; ++r) {
      if (!guarded || (tile * 16 + 8 * half + r) < nNodes) {
        float v = acc[r] + bv;
        po[r * C] = v > 0.0f ? v : 0.0f;
      }
    }
  }
}

__global__ void node_linear_wmma(const float* __restrict__ agg,
                                 const float* __restrict__ xin,
                                 const float* __restrict__ Wrel,
                                 const float* __restrict__ Wroot,
                                 const float* __restrict__ bias,
                                 float* __restrict__ xout,
                                 int nNodes) {
  const int lane  = threadIdx.x & 31;
  const int wave  = threadIdx.x >> 5;
  const int wpb   = blockDim.x >> 5;
  const int gwave = blockIdx.x * wpb + wave;
  const int nWav  = gridDim.x * wpb;

  const int half = lane >> 4;   // 0: lanes 0-15, 1: lanes 16-31
  const int nidx = lane & 15;   // A row (M) / B,D column (N)

  // ---- hoisted B fragments: Wcat(k,n), k<10 -> Wrel, 10<=k<20 -> Wroot ----
  v2f bf[5];
#pragma unroll
  for (int s = 0; s < 5; ++s) {
    const int k0 = 4 * s + 2 * half;        // even; pair stays in one matrix
    float b0 = 0.f, b1 = 0.f;
    if (nidx < C) {                          // once per wave: divergence OK
      const float* W0 = (k0 < C) ? (Wrel + (size_t)k0 * C)
                                 : (Wroot + (size_t)(k0 - C) * C);
      b0 = W0[nidx];
      b1 = W0[C + nidx];                     // row k0+1 of the same matrix
    }
    bf[s].x = b0;
    bf[s].y = b1;
  }
  const float bv = (nidx < C) ? bias[nidx] : 0.0f;

  const int fullTiles = nNodes >> 4;         // unguarded fast path
  for (int tile = gwave; tile < fullTiles; tile += nWav)
    node_tile(false, tile, nNodes, half, nidx, agg, xin, bf, bv, xout);

  // At most one partial tile; same grid-stride ownership rule.
  if (nNodes & 15) {
    if ((fullTiles % nWav) == gwave)
      node_tile(true, fullTiles, nNodes, half, nidx, agg, xin, bf, bv, xout);
  }
}

// ------------------------------------------------------------- linear head --
__global__ void final_linear(const float* __restrict__ x,
                             const float* __restrict__ Wl,
                             const float* __restrict__ bl,
                             float* __restrict__ out, int n) {
  int i = blockIdx.x * blockDim.x + threadIdx.x;
  if (i >= n) return;
  float s = bl[0];
  const float* xr = x + (size_t)i * C;
#pragma unroll
  for (int j = 0; j < 5; ++j) {
    float2 v = *reinterpret_cast<const float2*>(xr + 2 * j);
    s += v.x * Wl[2 * j] + v.y * Wl[2 * j + 1];
  }
  out[i] = s;
}

// ---------------------------------------------------------------- launcher --
extern "C" void kernel_launch(void* const* d_in, const int* in_sizes, int n_in,
                              void* d_out, int out_size, void* d_ws, size_t ws_size,
                              hipStream_t stream) {
  const float* x0  = (const float*)d_in[0];
  const int*   ei  = (const int*)  d_in[1];   // [2, E]
  const float* ew  = (const float*)d_in[2];
  const float* W1r = (const float*)d_in[3];
  const float* b1  = (const float*)d_in[4];
  const float* W1o = (const float*)d_in[5];
  const float* W2r = (const float*)d_in[6];
  const float* b2  = (const float*)d_in[7];
  const float* W2o = (const float*)d_in[8];
  const float* Wl  = (const float*)d_in[9];
  const float* bl  = (const float*)d_in[10];
  float* out = (float*)d_out;

  const int E = in_sizes[2];       // edge_weights element count
  const int n = in_sizes[0] / C;   // node count

  char*  ws  = (char*)d_ws;
  size_t per = (((size_t)n * C * sizeof(float)) + 255) & ~(size_t)255;
  float* agg = (float*)(ws);
  float* x1  = (float*)(ws + per);
  float* x2  = (float*)(ws + 2 * per);

  const int nf    = n * C;
  const int n4    = (nf + 3) / 4;            // ws rows are 256B-aligned
  const int zgrid = (n4 + 255) / 256;
  const int egrid = (E + 255) / 256;
  const int ngrid = 128;                      // 128 blocks x 8 waves = 1024
                                              // persistent waves over tiles

  // Layer 1
  zero_f32v4      <<<zgrid, 256, 0, stream>>>((float4*)agg, n4);
  edge_scatter    <<<egrid, 256, 0, stream>>>(x0, ei, ew, agg, E);
  node_linear_wmma<<<ngrid, 256, 0, stream>>>(agg, x0, W1r, W1o, b1, x1, n);
  // Layer 2
  zero_f32v4      <<<zgrid, 256, 0, stream>>>((float4*)agg, n4);
  edge_scatter    <<<egrid, 256, 0, stream>>>(x1, ei, ew, agg, E);
  node_linear_wmma<<<ngrid, 256, 0, stream>>>(agg, x1, W2r, W2o, b2, x2, n);
  // Head
  final_linear    <<<(n + 255) / 256, 256, 0, stream>>>(x2, Wl, bl, out, n);
}